// RNNDetector_43791486550106
// MI455X (gfx1250) — compile-verified
//
#include <hip/hip_runtime.h>

typedef __attribute__((ext_vector_type(2))) float v2f;
typedef __attribute__((ext_vector_type(8))) float v8f;

__device__ __forceinline__ float sigmoid_fast(float x) {
    return 1.0f / (1.0f + __expf(-x));
}
__device__ __forceinline__ float tanh_fast(float x) {
    // tanh(x) = 2*sigmoid(2x) - 1
    return 2.0f / (1.0f + __expf(-2.0f * x)) - 1.0f;
}

// ---------------------------------------------------------------------------
// Sequential LSTM scan: one workgroup, 256 threads (8 wave32s).
// Thread j owns gate-row j of W_hh (64 floats in VGPRs). h lives in LDS.
// LAYER0: input contribution is y[t]*W_ih0[j] + (b_ih[j]+b_hh[j]).
// else:   input contribution precomputed in xp (bias already folded).
// ---------------------------------------------------------------------------
template <bool LAYER0>
__global__ __launch_bounds__(256, 1) void lstm_scan_kernel(
    const float* __restrict__ src,   // LAYER0: y[T]; else xp[T*256]
    const float* __restrict__ W_hh,  // [256,64]
    const float* __restrict__ W_ih0, // [256] (LAYER0 only)
    const float* __restrict__ b_ih,  // [256] (LAYER0 only)
    const float* __restrict__ b_hh,  // [256] (LAYER0 only)
    float* __restrict__ h_out,       // [T,64]
    int T)
{
    __shared__ __align__(16) float h_sh[64];
    __shared__ float gates_sh[256];
    const int j = threadIdx.x;

    float w[64];
#pragma unroll
    for (int k = 0; k < 64; ++k) w[k] = W_hh[j * 64 + k];

    float wi = 0.0f, bj = 0.0f;
    if (LAYER0) { wi = W_ih0[j]; bj = b_ih[j] + b_hh[j]; }

    float c = 0.0f;
    if (j < 64) h_sh[j] = 0.0f;
    __syncthreads();

    for (int t = 0; t < T; ++t) {
        float g;
        if (LAYER0) {
            g = fmaf(src[t], wi, bj);
        } else {
            g = src[(size_t)t * 256 + j];
            if (t + 16 < T)
                __builtin_prefetch(&src[(size_t)(t + 16) * 256 + j], 0, 0);
        }
#pragma unroll
        for (int k = 0; k < 64; k += 4) {
            float4 hv = *(const float4*)&h_sh[k];   // LDS broadcast reads
            g = fmaf(w[k + 0], hv.x, g);
            g = fmaf(w[k + 1], hv.y, g);
            g = fmaf(w[k + 2], hv.z, g);
            g = fmaf(w[k + 3], hv.w, g);
        }
        gates_sh[j] = g;
        __syncthreads();
        if (j < 64) {
            float ig = sigmoid_fast(gates_sh[j]);
            float fg = sigmoid_fast(gates_sh[64 + j]);
            float gg = tanh_fast(gates_sh[128 + j]);
            float og = sigmoid_fast(gates_sh[192 + j]);
            c = fmaf(fg, c, ig * gg);
            float h = og * tanh_fast(c);
            h_sh[j] = h;
            h_out[(size_t)t * 64 + j] = h;
        }
        __syncthreads();
    }
}

// ---------------------------------------------------------------------------
// Fallback fused layer-1 scan (used only if workspace can't hold xp1):
// computes the input projection inline (128 FMAs/step instead of 64).
// ---------------------------------------------------------------------------
__global__ __launch_bounds__(256, 1) void lstm_scan_fused1_kernel(
    const float* __restrict__ h1,    // [T,64]
    const float* __restrict__ W_ih,  // [256,64]
    const float* __restrict__ W_hh,  // [256,64]
    const float* __restrict__ b_ih,  // [256]
    const float* __restrict__ b_hh,  // [256]
    float* __restrict__ h_out,       // [T,64]
    int T)
{
    __shared__ __align__(16) float h_sh[64];
    __shared__ __align__(16) float x_sh[64];
    __shared__ float gates_sh[256];
    const int j = threadIdx.x;

    float wi[64], w[64];
#pragma unroll
    for (int k = 0; k < 64; ++k) {
        wi[k] = W_ih[j * 64 + k];
        w[k]  = W_hh[j * 64 + k];
    }
    const float bj = b_ih[j] + b_hh[j];

    float c = 0.0f;
    if (j < 64) h_sh[j] = 0.0f;
    __syncthreads();

    for (int t = 0; t < T; ++t) {
        if (j < 64) x_sh[j] = h1[(size_t)t * 64 + j];
        __syncthreads();
        float g = bj;
#pragma unroll
        for (int k = 0; k < 64; k += 4) {
            float4 xv = *(const float4*)&x_sh[k];
            float4 hv = *(const float4*)&h_sh[k];
            g = fmaf(wi[k + 0], xv.x, g);
            g = fmaf(wi[k + 1], xv.y, g);
            g = fmaf(wi[k + 2], xv.z, g);
            g = fmaf(wi[k + 3], xv.w, g);
            g = fmaf(w[k + 0], hv.x, g);
            g = fmaf(w[k + 1], hv.y, g);
            g = fmaf(w[k + 2], hv.z, g);
            g = fmaf(w[k + 3], hv.w, g);
        }
        gates_sh[j] = g;
        __syncthreads();
        if (j < 64) {
            float ig = sigmoid_fast(gates_sh[j]);
            float fg = sigmoid_fast(gates_sh[64 + j]);
            float gg = tanh_fast(gates_sh[128 + j]);
            float og = sigmoid_fast(gates_sh[192 + j]);
            c = fmaf(fg, c, ig * gg);
            float h = og * tanh_fast(c);
            h_sh[j] = h;
            h_out[(size_t)t * 64 + j] = h;
        }
        __syncthreads();
    }
}

// ---------------------------------------------------------------------------
// xp1[T,256] = h1[T,64] @ W_ih1[256,64]^T + (b_ih1+b_hh1)
// WMMA f32 16x16x4: M=time tile, N=gate tile, K=64 in 16 chunks of 4.
// A lane layout: m = lane%16, a[v] = A[m][2*(lane/16)+v]  (v2f contiguous load)
// B lane layout: n = lane%16, b[v] = B[2*(lane/16)+v][n] = W[n*64 + k] (contiguous)
// C/D:           m = v + 8*(lane/16), n = lane%16
// ---------------------------------------------------------------------------
__global__ __launch_bounds__(256, 1) void xp1_gemm_kernel(
    const float* __restrict__ h1,    // [T,64]
    const float* __restrict__ W,     // [256,64]
    const float* __restrict__ b_ih,  // [256]
    const float* __restrict__ b_hh,  // [256]
    float* __restrict__ xp1,         // [T,256]
    int T)
{
    const int lane = threadIdx.x & 31;
    const int wave = threadIdx.x >> 5;
    const int tile = blockIdx.x * 8 + wave;
    const int tile_m = tile >> 4;       // 16 gate tiles per time tile
    const int tile_n = tile & 15;
    const int t0 = tile_m * 16;
    if (t0 >= T) return;
    const int g0 = tile_n * 16;
    const int half = lane >> 4;
    const int mr = lane & 15;

    const float bias = b_ih[g0 + mr] + b_hh[g0 + mr];
    v8f acc;
#pragma unroll
    for (int v = 0; v < 8; ++v) acc[v] = bias;

    const float* arow = &h1[(size_t)(t0 + mr) * 64 + half * 2];
    const float* brow = &W[(size_t)(g0 + mr) * 64 + half * 2];
#pragma unroll
    for (int kc = 0; kc < 16; ++kc) {
        v2f a = *(const v2f*)(arow + kc * 4);
        v2f b = *(const v2f*)(brow + kc * 4);
        acc = __builtin_amdgcn_wmma_f32_16x16x4_f32(
            false, a, false, b, (short)0, acc, false, false);
    }
#pragma unroll
    for (int v = 0; v < 8; ++v)
        xp1[(size_t)(t0 + half * 8 + v) * 256 + g0 + mr] = acc[v];
}

// ---------------------------------------------------------------------------
// out[T,16] = h2[T,64] @ W_lin[16,64]^T + b_lin  — single N tile, WMMA f32.
// ---------------------------------------------------------------------------
__global__ __launch_bounds__(256, 1) void out_gemm_kernel(
    const float* __restrict__ h2,    // [T,64]
    const float* __restrict__ W_lin, // [16,64]
    const float* __restrict__ b_lin, // [16]
    float* __restrict__ out,         // [T,16]
    int T)
{
    const int lane = threadIdx.x & 31;
    const int wave = threadIdx.x >> 5;
    const int tile_m = blockIdx.x * 8 + wave;
    const int t0 = tile_m * 16;
    if (t0 >= T) return;
    const int half = lane >> 4;
    const int mr = lane & 15;

    const float bias = b_lin[mr];
    v8f acc;
#pragma unroll
    for (int v = 0; v < 8; ++v) acc[v] = bias;

    const float* arow = &h2[(size_t)(t0 + mr) * 64 + half * 2];
    const float* brow = &W_lin[(size_t)mr * 64 + half * 2];
#pragma unroll
    for (int kc = 0; kc < 16; ++kc) {
        v2f a = *(const v2f*)(arow + kc * 4);
        v2f b = *(const v2f*)(brow + kc * 4);
        acc = __builtin_amdgcn_wmma_f32_16x16x4_f32(
            false, a, false, b, (short)0, acc, false, false);
    }
#pragma unroll
    for (int v = 0; v < 8; ++v)
        out[(size_t)(t0 + half * 8 + v) * 16 + mr] = acc[v];
}

extern "C" void kernel_launch(void* const* d_in, const int* in_sizes, int n_in,
                              void* d_out, int out_size, void* d_ws, size_t ws_size,
                              hipStream_t stream)
{
    const float* y     = (const float*)d_in[0];
    const float* W_ih0 = (const float*)d_in[1];
    const float* W_hh0 = (const float*)d_in[2];
    const float* b_ih0 = (const float*)d_in[3];
    const float* b_hh0 = (const float*)d_in[4];
    const float* W_ih1 = (const float*)d_in[5];
    const float* W_hh1 = (const float*)d_in[6];
    const float* b_ih1 = (const float*)d_in[7];
    const float* b_hh1 = (const float*)d_in[8];
    const float* W_lin = (const float*)d_in[9];
    const float* b_lin = (const float*)d_in[10];
    float* out = (float*)d_out;
    const int T = in_sizes[0];      // 262144

    const size_t hbytes  = (size_t)T * 64 * sizeof(float);   // 64 MB
    const size_t xpbytes = (size_t)T * 256 * sizeof(float);  // 256 MB

    float* h1 = (float*)d_ws;

    // Layer 0 scan (serial critical path: 64 FMAs + 5 transcendentals/step).
    lstm_scan_kernel<true><<<1, 256, 0, stream>>>(
        y, W_hh0, W_ih0, b_ih0, b_hh0, h1, T);

    float* h2;
    if (ws_size >= hbytes + xpbytes) {
        // WMMA path: precompute xp1, keep the serial scan minimal.
        float* xp1 = (float*)((char*)d_ws + hbytes);
        xp1_gemm_kernel<<<T / 8, 256, 0, stream>>>(
            h1, W_ih1, b_ih1, b_hh1, xp1, T);
        h2 = h1;  // h1 region is dead once xp1 exists; reuse for h2
        lstm_scan_kernel<false><<<1, 256, 0, stream>>>(
            xp1, W_hh1, nullptr, nullptr, nullptr, h2, T);
    } else {
        // Fused fallback: input projection computed inside the scan.
        h2 = (float*)((char*)d_ws + hbytes);
        lstm_scan_fused1_kernel<<<1, 256, 0, stream>>>(
            h1, W_ih1, W_hh1, b_ih1, b_hh1, h2, T);
    }

    // Output projection (one 16-wide N tile), WMMA f32.
    out_gemm_kernel<<<T / 128, 256, 0, stream>>>(h2, W_lin, b_lin, out, T);
}